// ParcelRebalancedSoftmaxLoss_13477607375413
// MI455X (gfx1250) — compile-verified
//
#include <hip/hip_runtime.h>

// Problem constants (from the reference)
#define NUM_CLASSES 20
#define NUM_PARCELS 8192
#define IGNORE_INDEX 255
#define NB 8
#define HW (512 * 512)
#define M_TOT (NB * HW)   // 2,097,152 pixels

typedef __attribute__((ext_vector_type(2))) float v2f;
typedef __attribute__((ext_vector_type(8))) float v8f;

// ---------------------------------------------------------------------------
// Kernel 0: initialize accumulators in workspace (must re-run every call).
// ---------------------------------------------------------------------------
__global__ void prsl_init_kernel(float* __restrict__ seg, float* __restrict__ cnt,
                                 int* __restrict__ first) {
    int i = blockIdx.x * blockDim.x + threadIdx.x;
    if (i < NUM_PARCELS * NUM_CLASSES) seg[i] = 0.0f;
    if (i < NUM_PARCELS) {
        cnt[i] = 0.0f;
        first[i] = M_TOT;   // "M" sentinel, matches jnp.where(valid, idx, M)
    }
}

// ---------------------------------------------------------------------------
// Kernel 1: scatter-add logits into per-parcel sums (L2-resident accumulators).
// One thread per pixel; pred reads are coalesced per channel plane (NCHW).
// ---------------------------------------------------------------------------
__global__ void prsl_scatter_kernel(const float* __restrict__ pred,
                                    const int* __restrict__ tgt,
                                    const int* __restrict__ par,
                                    float* __restrict__ seg,
                                    float* __restrict__ cnt,
                                    int* __restrict__ first) {
    int m = blockIdx.x * blockDim.x + threadIdx.x;
    if (m >= M_TOT) return;
    int t = tgt[m];
    int p = par[m];
    if (t == IGNORE_INDEX) return;

    atomicAdd(&cnt[p], 1.0f);
    atomicMin(&first[p], m);

    int n  = m / HW;
    int hw = m - n * HW;
    const float* base = pred + (size_t)n * NUM_CLASSES * HW + hw;
    float* s = seg + (size_t)p * NUM_CLASSES;
#pragma unroll
    for (int c = 0; c < NUM_CLASSES; ++c) {
        atomicAdd(&s[c], base[(size_t)c * HW]);
    }
}

// ---------------------------------------------------------------------------
// Kernel 2: per-parcel balanced log-softmax CE.
// One wave32 handles 16 parcels. Row-sum of exp() done on the matrix core:
// 5 chained V_WMMA_F32_16X16X4_F32 with B = ones (full f32, no precision loss
// beyond normal fp32 accumulation). Row max + final NLL on VALU.
// ---------------------------------------------------------------------------
__global__ void prsl_parcel_loss_kernel(const float* __restrict__ seg,
                                        const float* __restrict__ cnt,
                                        const int* __restrict__ first,
                                        const int* __restrict__ tgt,
                                        const int* __restrict__ cls,
                                        float* __restrict__ nll,
                                        float* __restrict__ pres) {
    __shared__ float shlog[NUM_CLASSES];
    __shared__ float red[8 * 16];   // 8 waves/block x 16 rows

    int tid = threadIdx.x;
    if (tid < NUM_CLASSES) shlog[tid] = __logf((float)cls[tid]);
    __syncthreads();

    int lane = tid & 31;
    int wv   = tid >> 5;            // wave index within block (0..7)
    int half = lane >> 4;           // 0: K=0,1 lanes ; 1: K=2,3 lanes
    int row  = lane & 15;           // matrix row = parcel within this wave
    int p    = (blockIdx.x * 8 + wv) * 16 + row;

    float cp  = cnt[p];
    float inv = 1.0f / fmaxf(cp, 1.0f);
    const float* sp = seg + (size_t)p * NUM_CLASSES;

    // Row max over the 20 balanced logits (both half-waves compute the same
    // value for their shared row -> identical A inputs).
    float mx = -1e30f;
#pragma unroll
    for (int c = 0; c < NUM_CLASSES; ++c)
        mx = fmaxf(mx, sp[c] * inv + shlog[c]);

    // Row-sum of exp(bal - mx) via WMMA: A(16x4 f32) x ones(4x16) accumulated
    // over 5 K-chunks (20 columns total). A layout per ISA 7.12.2:
    //   lanes 0-15: v0=K0, v1=K1 ; lanes 16-31: v0=K2, v1=K3.
    v2f bones; bones.x = 1.0f; bones.y = 1.0f;
    v8f acc = {};
#pragma unroll
    for (int i = 0; i < 5; ++i) {
        int c0 = 4 * i + 2 * half;
        v2f a;
        a.x = __expf(sp[c0]     * inv + shlog[c0]     - mx);
        a.y = __expf(sp[c0 + 1] * inv + shlog[c0 + 1] - mx);
        acc = __builtin_amdgcn_wmma_f32_16x16x4_f32(
            /*neg_a=*/false, a, /*neg_b=*/false, bones,
            /*c_mod=*/(short)0, acc, /*reuse_a=*/false, /*reuse_b=*/false);
    }

    // D layout: VGPR j holds row j (lanes 0-15) / row 8+j (lanes 16-31);
    // every column is identical (B was all ones). Stage 16 row-sums via LDS.
    if (lane == 0 || lane == 16) {
        int rbase = wv * 16 + half * 8;
#pragma unroll
        for (int j = 0; j < 8; ++j) red[rbase + j] = acc[j];
    }
    __syncthreads();
    float rsum = red[wv * 16 + row];

    if (half == 0) {   // lanes 0..15 write the 16 parcels of this wave
        int idx = min(first[p], M_TOT - 1);
        int lbl = tgt[idx];
        if (lbl < 0 || lbl >= NUM_CLASSES) lbl = 0;  // clamp (ignore-label rows are zeroed anyway)
        float bal_l = sp[lbl] * inv + shlog[lbl];
        float v = __logf(rsum) + mx - bal_l;          // -log_softmax at label
        bool present = cp > 0.0f;
        nll[p]  = present ? v : 0.0f;
        pres[p] = present ? 1.0f : 0.0f;
    }
}

// ---------------------------------------------------------------------------
// Kernel 3: deterministic 8192 -> 1 reduction + final division.
// ---------------------------------------------------------------------------
__global__ void prsl_reduce_kernel(const float* __restrict__ nll,
                                   const float* __restrict__ pres,
                                   float* __restrict__ out) {
    __shared__ float s1[256];
    __shared__ float s2[256];
    int tid = threadIdx.x;
    float a = 0.0f, b = 0.0f;
    for (int i = tid; i < NUM_PARCELS; i += 256) { a += nll[i]; b += pres[i]; }
    s1[tid] = a; s2[tid] = b;
    __syncthreads();
    for (int st = 128; st > 0; st >>= 1) {
        if (tid < st) { s1[tid] += s1[tid + st]; s2[tid] += s2[tid + st]; }
        __syncthreads();
    }
    if (tid == 0) out[0] = s1[0] / fmaxf(s2[0], 1.0f);
}

// ---------------------------------------------------------------------------
// Launcher. Workspace layout (768 KB total):
//   seg  : float[8192*20]  @ 0
//   cnt  : float[8192]     @ 655360
//   first: int  [8192]     @ 688128
//   nll  : float[8192]     @ 720896
//   pres : float[8192]     @ 753664
// ---------------------------------------------------------------------------
extern "C" void kernel_launch(void* const* d_in, const int* in_sizes, int n_in,
                              void* d_out, int out_size, void* d_ws, size_t ws_size,
                              hipStream_t stream) {
    const float* pred = (const float*)d_in[0];
    const int*   tgt  = (const int*)d_in[1];
    const int*   par  = (const int*)d_in[2];
    const int*   cls  = (const int*)d_in[3];
    float* out = (float*)d_out;

    char* ws = (char*)d_ws;
    float* seg   = (float*)(ws);
    float* cnt   = (float*)(ws + (size_t)NUM_PARCELS * NUM_CLASSES * 4);
    int*   first = (int*)  (ws + (size_t)NUM_PARCELS * (NUM_CLASSES + 1) * 4);
    float* nll   = (float*)(ws + (size_t)NUM_PARCELS * (NUM_CLASSES + 2) * 4);
    float* pres  = (float*)(ws + (size_t)NUM_PARCELS * (NUM_CLASSES + 3) * 4);

    hipLaunchKernelGGL(prsl_init_kernel,
                       dim3((NUM_PARCELS * NUM_CLASSES + 255) / 256), dim3(256), 0, stream,
                       seg, cnt, first);

    hipLaunchKernelGGL(prsl_scatter_kernel,
                       dim3((M_TOT + 255) / 256), dim3(256), 0, stream,
                       pred, tgt, par, seg, cnt, first);

    // 8192 parcels / (16 per wave * 8 waves per block) = 64 blocks
    hipLaunchKernelGGL(prsl_parcel_loss_kernel,
                       dim3(NUM_PARCELS / 128), dim3(256), 0, stream,
                       seg, cnt, first, tgt, cls, nll, pres);

    hipLaunchKernelGGL(prsl_reduce_kernel, dim3(1), dim3(256), 0, stream,
                       nll, pres, out);
}